// HQCNN_Ang_noQP_58660663329364
// MI455X (gfx1250) — compile-verified
//
#include <hip/hip_runtime.h>
#include <math.h>

typedef __attribute__((ext_vector_type(2))) float v2f;
typedef __attribute__((ext_vector_type(8))) float v8f;

// ---------------- complex helpers (precompute kernel only) ----------------
struct c2 { float r, i; };
__device__ __forceinline__ c2 cmul(c2 a, c2 b){ return c2{a.r*b.r - a.i*b.i, a.r*b.i + a.i*b.r}; }
__device__ __forceinline__ c2 cadd(c2 a, c2 b){ return c2{a.r+b.r, a.i+b.i}; }

__device__ void u3gate(float th, float ph, float lam, c2 g[2][2]) {
  float ct = cosf(0.5f*th), st = sinf(0.5f*th);
  g[0][0] = c2{ct, 0.f};
  g[0][1] = c2{-cosf(lam)*st, -sinf(lam)*st};
  g[1][0] = c2{cosf(ph)*st, sinf(ph)*st};
  g[1][1] = c2{cosf(ph+lam)*ct, sinf(ph+lam)*ct};
}
__device__ void rygate(float t, c2 g[2][2]) {
  float c = cosf(0.5f*t), s = sinf(0.5f*t);
  g[0][0]=c2{c,0.f}; g[0][1]=c2{-s,0.f}; g[1][0]=c2{s,0.f}; g[1][1]=c2{c,0.f};
}
__device__ void rzgate(float t, c2 g[2][2]) {
  g[0][0]=c2{cosf(0.5f*t), -sinf(0.5f*t)}; g[0][1]=c2{0.f,0.f};
  g[1][0]=c2{0.f,0.f};  g[1][1]=c2{cosf(0.5f*t),  sinf(0.5f*t)};
}
// A = kron(g0,g1) @ A   (4x4 complex)
__device__ void kron_mul(c2 g0[2][2], c2 g1[2][2], c2 A[4][4]) {
  c2 G[4][4];
  for (int i=0;i<2;i++) for(int j=0;j<2;j++)
    for (int a=0;a<2;a++) for(int b=0;b<2;b++)
      G[i*2+j][a*2+b] = cmul(g0[i][a], g1[j][b]);
  c2 T[4][4];
  for (int i=0;i<4;i++) for (int j=0;j<4;j++) {
    c2 s{0.f,0.f};
    for (int k=0;k<4;k++) s = cadd(s, cmul(G[i][k], A[k][j]));
    T[i][j] = s;
  }
  for (int i=0;i<4;i++) for(int j=0;j<4;j++) A[i][j]=T[i][j];
}
__device__ void swap_rows(c2 A[4][4], int r0, int r1){
  for (int j=0;j<4;j++){ c2 t=A[r0][j]; A[r0][j]=A[r1][j]; A[r1][j]=t; }
}

// ------------- Kernel 1: fold circuit + W2 into ten 8x8 matrices -------------
// Bm[c*64 + j*8 + jj] = sum_k s_ck*(ReM[k][j]ReM[k][jj] + ImM[k][j]ImM[k][jj])
// where M = circuit unitary restricted to input basis states 0..7,
// s_ck = sum_i W2[c,i] * (1 - 2*bit_{7-i}(k)).
__global__ __launch_bounds__(256) void hqcnn_precompute(
    const float* __restrict__ qw0, const float* __restrict__ qw1,
    const float* __restrict__ qw2, const float* __restrict__ qw3,
    const float* __restrict__ qw4, const float* __restrict__ qw5,
    const float* __restrict__ qw6, const float* __restrict__ W2,
    float* __restrict__ Bm)
{
  __shared__ float Sre[256][8];
  __shared__ float Sim[256][8];
  __shared__ float sS[256][10];
  const int tid = threadIdx.x;

  // Uniform per-thread build of the SU4 unitary (gate order right-to-left).
  c2 U[4][4];
  for (int i=0;i<4;i++) for (int j=0;j<4;j++) U[i][j] = c2{(i==j)?1.f:0.f, 0.f};
  c2 g0[2][2], g1[2][2];
  u3gate(qw0[0], qw0[1], qw0[2], g0);
  u3gate(qw1[0], qw1[1], qw1[2], g1);
  kron_mul(g0, g1, U);
  swap_rows(U, 2, 3);                       // CNOT (control = first qubit)
  rygate(qw2[0], g0); rzgate(qw3[0], g1);
  kron_mul(g0, g1, U);
  swap_rows(U, 1, 3);                       // CNOT (control = second qubit)
  rygate(qw4[0], g0);
  g1[0][0]=c2{1.f,0.f}; g1[0][1]=c2{0.f,0.f};
  g1[1][0]=c2{0.f,0.f}; g1[1][1]=c2{1.f,0.f};
  kron_mul(g0, g1, U);
  swap_rows(U, 2, 3);                       // CNOT (control = first qubit)
  u3gate(qw5[0], qw5[1], qw5[2], g0);
  u3gate(qw6[0], qw6[1], qw6[2], g1);
  kron_mul(g0, g1, U);

  // State matrix columns = basis states e_0..e_7
  for (int j=0;j<8;j++){ Sre[tid][j] = (tid==j)?1.f:0.f; Sim[tid][j] = 0.f; }

  const int qa[8] = {0,2,4,6,1,3,5,7};
  const int qb[8] = {1,3,5,7,2,4,6,0};
  for (int g = 0; g < 8; g++) {
    const int pa = 7 - qa[g], pb = 7 - qb[g];   // qubit 0 = MSB of state index
    const int ma = 1<<pa, mb = 1<<pb;
    __syncthreads();
    for (int task = tid; task < 512; task += 256) {
      const int col = task & 7, grp = task >> 3;
      int k00 = 0, gb = 0;
      for (int bit=0; bit<8; bit++){
        if (bit==pa || bit==pb) continue;
        k00 |= ((grp >> gb) & 1) << bit;
        gb++;
      }
      const int kk[4] = {k00, k00|mb, k00|ma, k00|ma|mb};
      c2 s[4], n[4];
      for (int q=0;q<4;q++) s[q] = c2{Sre[kk[q]][col], Sim[kk[q]][col]};
      for (int i2=0;i2<4;i2++){
        c2 acc{0.f,0.f};
        for (int q=0;q<4;q++) acc = cadd(acc, cmul(U[i2][q], s[q]));
        n[i2] = acc;
      }
      for (int q=0;q<4;q++){ Sre[kk[q]][col] = n[q].r; Sim[kk[q]][col] = n[q].i; }
    }
  }
  __syncthreads();

  // sign-folded W2
  {
    const int k = tid;
    for (int c=0;c<10;c++){
      float s = 0.f;
      for (int i=0;i<8;i++){
        float sgn = ((k >> (7-i)) & 1) ? -1.f : 1.f;
        s += W2[c*8+i] * sgn;
      }
      sS[k][c] = s;
    }
  }
  __syncthreads();

  for (int e = tid; e < 640; e += 256) {
    const int c = e >> 6, j = (e>>3)&7, jj = e&7;
    float acc = 0.f;
    for (int k=0;k<256;k++)
      acc += sS[k][c]*(Sre[k][j]*Sre[k][jj] + Sim[k][j]*Sim[k][jj]);
    Bm[e] = acc;
  }
}

// ------------- Kernel 2: WMMA GEMM + fused normalized quadratic forms -------------
// Block = 128 threads = 4 waves; each wave computes TWO 16-row tiles sharing one
// B fragment per K-step (V_WMMA_F32_16X16X4_F32). All loads are unconditional:
// lanes with N>=8 load duplicate W1 rows; duplicate D columns are never stored.
__global__ __launch_bounds__(128) void hqcnn_main(
    const float* __restrict__ x, const float* __restrict__ W1,
    const float* __restrict__ b1, const float* __restrict__ b2,
    const float* __restrict__ Bm, float* __restrict__ out, int F)
{
  __shared__ float sBm[640];
  __shared__ float sH[128*9];
  const int tid = threadIdx.x;
  for (int i = tid; i < 640; i += 128) sBm[i] = Bm[i];

  const int wave = tid >> 5;       // 0..3, each wave owns 32 rows
  const int lane = tid & 31;
  const int l15  = lane & 15;
  const int hi   = lane >> 4;      // 0: K={0,1}, 1: K={2,3} per ISA f32 A/B layout

  const long rbase = (long)blockIdx.x*128 + wave*32;
  const float* __restrict__ aptr0 = x  + (rbase      + l15) * (long)F + 2*hi;
  const float* __restrict__ aptr1 = x  + (rbase + 16 + l15) * (long)F + 2*hi;
  const float* __restrict__ bptr  = W1 + (long)(l15 & 7) * (long)F + 2*hi;

  v8f acc0 = {}, acc1 = {};
  #pragma unroll 4
  for (int k = 0; k < F; k += 4) {
    v2f b  = *(const v2f*)(bptr  + k);
    v2f a0 = *(const v2f*)(aptr0 + k);
    v2f a1 = *(const v2f*)(aptr1 + k);
    acc0 = __builtin_amdgcn_wmma_f32_16x16x4_f32(
        false, a0, false, b, (short)0, acc0, false, false);
    acc1 = __builtin_amdgcn_wmma_f32_16x16x4_f32(
        false, a1, false, b, (short)0, acc1, false, false);
  }

  // D layout: VGPR v -> M = v + 8*hi, N = l15. Keep N<8 columns only.
  if (l15 < 8) {
    #pragma unroll
    for (int v = 0; v < 8; v++) {
      const int r0 = wave*32 + v + 8*hi;
      sH[r0*9 + l15]      = acc0[v];
      sH[(r0+16)*9 + l15] = acc1[v];
    }
  }
  __syncthreads();

  // One thread per row: normalize and evaluate ten 8x8 quadratic forms.
  {
    float p[8];
    float ss = 0.f;
    #pragma unroll
    for (int j=0;j<8;j++){ float v = sH[tid*9+j] + b1[j]; p[j]=v; ss += v*v; }
    const float inv = rsqrtf(ss);
    #pragma unroll
    for (int j=0;j<8;j++) p[j] *= inv;
    const long row = (long)blockIdx.x*128 + tid;
    #pragma unroll
    for (int c=0;c<10;c++){
      float o = b2[c];
      #pragma unroll
      for (int j=0;j<8;j++){
        const float pj = p[j];
        #pragma unroll
        for (int jj=0;jj<8;jj++) o += pj * p[jj] * sBm[c*64 + j*8 + jj];
      }
      out[row*10 + c] = o;
    }
  }
}

extern "C" void kernel_launch(void* const* d_in, const int* in_sizes, int n_in,
                              void* d_out, int out_size, void* d_ws, size_t ws_size,
                              hipStream_t stream) {
  const float* x   = (const float*)d_in[0];
  const float* W1  = (const float*)d_in[1];
  const float* b1  = (const float*)d_in[2];
  const float* qw0 = (const float*)d_in[3];
  const float* qw1 = (const float*)d_in[4];
  const float* qw2 = (const float*)d_in[5];
  const float* qw3 = (const float*)d_in[6];
  const float* qw4 = (const float*)d_in[7];
  const float* qw5 = (const float*)d_in[8];
  const float* qw6 = (const float*)d_in[9];
  const float* W2  = (const float*)d_in[10];
  const float* b2  = (const float*)d_in[11];
  float* out = (float*)d_out;
  float* Bm  = (float*)d_ws;   // 640 floats

  const int  F     = in_sizes[1] / 8;              // 2048
  const long Brows = (long)in_sizes[0] / F;        // 32768

  hqcnn_precompute<<<1, 256, 0, stream>>>(qw0, qw1, qw2, qw3, qw4, qw5, qw6, W2, Bm);

  const int blocks = (int)(Brows / 128);           // 256
  hqcnn_main<<<blocks, 128, 0, stream>>>(x, W1, b1, b2, Bm, out, F);
}